// Attention_49374944035385
// MI455X (gfx1250) — compile-verified
//
#include <hip/hip_runtime.h>
#include <math.h>

typedef __attribute__((ext_vector_type(2))) float v2f;
typedef __attribute__((ext_vector_type(8))) float v8f;

#define NUM_SEQS     32
#define NUM_HEADS    32
#define NUM_KV_HEADS 8
#define GQA          4
#define HEAD         128
#define BLK          16
#define MAXB         256
#define NWAVES       4
#define THREADS      (NWAVES * 32)

#define TSTRIDE 132               // padded row stride (floats): bank-conflict-free, 16B aligned
// LDS layout (float offsets)
#define QS_OFF  0                 // q^T 16 x 132 (rows >= 4 and cols >= 128 are zero)
#define OB_OFF  2112              // 4 x 128 output combine buffer
#define RM_OFF  2624              // NWAVES x 4 running max
#define RL_OFF  2640              // NWAVES x 4 running sum
#define WV_OFF  2656              // per-wave region
#define WV_SZ   (2 * 16 * TSTRIDE + 16 * 18)        // K tile + V tile + probs = 4512
#define LDS_FLOATS (WV_OFF + NWAVES * WV_SZ)        // 20704 floats
#define LDS_BYTES  (LDS_FLOATS * 4)                 // 82816 B (opt-in > 64KB)

#if __has_builtin(__builtin_amdgcn_global_load_async_to_lds_b128) && \
    __has_builtin(__builtin_amdgcn_s_wait_asynccnt)
#define USE_ASYNC 1
typedef __attribute__((__vector_size__(16))) int i128_t;
typedef __attribute__((address_space(1))) i128_t gi128_t;   // global 16B chunk
typedef __attribute__((address_space(3))) i128_t li128_t;   // LDS 16B chunk
#else
#define USE_ASYNC 0
#endif

__device__ __forceinline__ v8f wmma_f32(v2f a, v2f b, v8f c) {
  return __builtin_amdgcn_wmma_f32_16x16x4_f32(false, a, false, b, (short)0, c,
                                               false, false);
}

// Copy one 512B K/V row (HEAD floats) into an LDS tile row: 32 lanes x 16B.
__device__ __forceinline__ void stage_row(const float* __restrict__ src,
                                          float* __restrict__ dst, int lane) {
#if USE_ASYNC
  __builtin_amdgcn_global_load_async_to_lds_b128(
      (gi128_t*)(src + lane * 4), (li128_t*)(dst + lane * 4), 0, 0);
#else
  *(float4*)(dst + lane * 4) = *(const float4*)(src + lane * 4);
#endif
}

__global__ __launch_bounds__(THREADS) void paged_attn_decode(
    const float* __restrict__ q,
    const float* __restrict__ knew,
    const float* __restrict__ vnew,
    const float* __restrict__ kcache,
    const float* __restrict__ vcache,
    const int* __restrict__ btab,
    const int* __restrict__ ctx,
    float* __restrict__ out)
{
  extern __shared__ float smem[];
  const int tid  = threadIdx.x;
  const int lane = tid & 31;
  const int w    = tid >> 5;
  const int n    = lane & 15;     // token row (A) / head column (B,C,D)
  const int h    = lane >> 4;     // half-wave select
  const int tb   = h * 8;         // C/D row base held by this lane

  const int kvh = blockIdx.x;
  const int seq = blockIdx.y;
  const int len = ctx[seq];
  const int ntiles = (len + BLK - 1) / BLK;

  float* qs = smem + QS_OFF;
  float* ob = smem + OB_OFF;
  float* rm = smem + RM_OFF;
  float* rl = smem + RL_OFF;
  float* kb = smem + WV_OFF + w * WV_SZ;   // 16 x 132 K tile
  float* vb = kb + 16 * TSTRIDE;           // 16 x 132 V tile
  float* pt = vb + 16 * TSTRIDE;           // 16 x 18 probs [head][token]

  const float scale = 0.08838834764831845f;  // 1/sqrt(128)

  // q -> LDS, pre-scaled, zero-padded; zero the combine buffer.
  for (int idx = tid; idx < 16 * TSTRIDE; idx += THREADS) {
    int row = idx / TSTRIDE, col = idx - row * TSTRIDE;
    float val = 0.f;
    if (row < GQA && col < HEAD)
      val = q[((size_t)seq * NUM_HEADS + kvh * GQA + row) * HEAD + col] * scale;
    qs[idx] = val;
  }
  for (int idx = tid; idx < GQA * HEAD; idx += THREADS) ob[idx] = 0.f;
  __syncthreads();

  const float* krow_new = knew + ((size_t)seq * NUM_KV_HEADS + kvh) * HEAD;
  const float* vrow_new = vnew + ((size_t)seq * NUM_KV_HEADS + kvh) * HEAD;

  float m_run = -INFINITY, l_run = 0.f;
  v8f oacc[8];
  #pragma unroll
  for (int c = 0; c < 8; ++c) oacc[c] = v8f{};

  for (int tile = w; tile < ntiles; tile += NWAVES) {
    const int pb = btab[seq * MAXB + tile];
    const size_t blk_base =
        (size_t)pb * BLK * (size_t)(NUM_KV_HEADS * HEAD) + (size_t)kvh * HEAD;

    // --- stage K rows, then V rows (async completes in order: K first) ---
    #pragma unroll
    for (int r = 0; r < BLK; ++r) {
      int tg = tile * BLK + r;
      const float* ksrc = (tg == len - 1)
          ? krow_new : kcache + blk_base + (size_t)r * (NUM_KV_HEADS * HEAD);
      stage_row(ksrc, kb + r * TSTRIDE, lane);
    }
    #pragma unroll
    for (int r = 0; r < BLK; ++r) {
      int tg = tile * BLK + r;
      const float* vsrc = (tg == len - 1)
          ? vrow_new : vcache + blk_base + (size_t)r * (NUM_KV_HEADS * HEAD);
      stage_row(vsrc, vb + r * TSTRIDE, lane);
    }
#if USE_ASYNC
    __builtin_amdgcn_s_wait_asynccnt(16);   // 16 K rows landed; V still in flight
    asm volatile("" ::: "memory");
#endif

    // --- scores: S(16tok x 16head) = K_tile(16x128) x q^T(128x16), fp32 WMMA ---
    v8f s0 = v8f{}, s1 = v8f{};
    #pragma unroll
    for (int k0 = 0; k0 < HEAD; k0 += 8) {
      v2f a0 = *(const v2f*)(kb + n * TSTRIDE + k0 + 2 * h);
      v2f b0 = *(const v2f*)(qs + n * TSTRIDE + k0 + 2 * h);
      s0 = wmma_f32(a0, b0, s0);
      v2f a1 = *(const v2f*)(kb + n * TSTRIDE + k0 + 4 + 2 * h);
      v2f b1 = *(const v2f*)(qs + n * TSTRIDE + k0 + 4 + 2 * h);
      s1 = wmma_f32(a1, b1, s1);
    }
    v8f s = s0 + s1;

    // --- mask invalid tokens + online softmax (per head = per lane column) ---
    float sv[8];
    float smax = -INFINITY;
    #pragma unroll
    for (int j = 0; j < 8; ++j) {
      int tg = tile * BLK + tb + j;
      float x = (tg < len) ? s[j] : -INFINITY;
      sv[j] = x;
      smax = fmaxf(smax, x);
    }
    smax = fmaxf(smax, __shfl_xor(smax, 16, 32));
    float mnew = fmaxf(m_run, smax);
    float fac  = expf(m_run - mnew);
    #pragma unroll
    for (int c = 0; c < 8; ++c) oacc[c] = oacc[c] * fac;
    float psum = 0.f, pv[8];
    #pragma unroll
    for (int j = 0; j < 8; ++j) { float p = expf(sv[j] - mnew); pv[j] = p; psum += p; }
    psum += __shfl_xor(psum, 16, 32);
    l_run = l_run * fac + psum;
    m_run = mnew;

    // probs -> LDS transposed [head][token] (stride 18: 8B-aligned, conflict-free)
    #pragma unroll
    for (int j = 0; j < 8; ++j) pt[n * 18 + tb + j] = pv[j];

#if USE_ASYNC
    __builtin_amdgcn_s_wait_asynccnt(0);    // V rows landed
    asm volatile("" ::: "memory");
#endif

    // --- PV: outT(128x16head) += V^T(128 x 16tok) x P(16tok x 16head) ---
    #pragma unroll
    for (int t0 = 0; t0 < BLK; t0 += 4) {
      v2f bfrag = *(const v2f*)(pt + n * 18 + t0 + 2 * h);
      #pragma unroll
      for (int c = 0; c < 8; ++c) {
        v2f afrag;
        afrag.x = vb[(t0 + 2 * h)     * TSTRIDE + c * 16 + n];
        afrag.y = vb[(t0 + 2 * h + 1) * TSTRIDE + c * 16 + n];
        oacc[c] = wmma_f32(afrag, bfrag, oacc[c]);
      }
    }
  }

  // --- cross-wave flash combine ---
  if (lane < GQA) { rm[w * GQA + lane] = m_run; rl[w * GQA + lane] = l_run; }
  __syncthreads();

  int gg = (n < GQA) ? n : 0;
  float Mg = -INFINITY;
  #pragma unroll
  for (int ww = 0; ww < NWAVES; ++ww) Mg = fmaxf(Mg, rm[ww * GQA + gg]);
  float fac2 = expf(m_run - Mg);
  if (n < GQA) {
    #pragma unroll
    for (int c = 0; c < 8; ++c) {
      #pragma unroll
      for (int j = 0; j < 8; ++j)
        atomicAdd(&ob[n * HEAD + c * 16 + tb + j], fac2 * oacc[c][j]);
    }
  }
  __syncthreads();

  for (int idx = tid; idx < GQA * HEAD; idx += THREADS) {
    int g = idx >> 7, d = idx & 127;
    float M2 = -INFINITY;
    #pragma unroll
    for (int ww = 0; ww < NWAVES; ++ww) M2 = fmaxf(M2, rm[ww * GQA + g]);
    float ls = 0.f;
    #pragma unroll
    for (int ww = 0; ww < NWAVES; ++ww)
      ls += expf(rm[ww * GQA + g] - M2) * rl[ww * GQA + g];
    out[((size_t)seq * NUM_HEADS + kvh * GQA + g) * HEAD + d] = ob[idx] / ls;
  }
}

extern "C" void kernel_launch(void* const* d_in, const int* in_sizes, int n_in,
                              void* d_out, int out_size, void* d_ws, size_t ws_size,
                              hipStream_t stream) {
  (void)in_sizes; (void)n_in; (void)out_size; (void)d_ws; (void)ws_size;
  const float* q  = (const float*)d_in[0];
  const float* k  = (const float*)d_in[1];
  const float* v  = (const float*)d_in[2];
  const float* kc = (const float*)d_in[3];
  const float* vc = (const float*)d_in[4];
  const int*   bt = (const int*)d_in[5];
  const int*   cl = (const int*)d_in[6];
  // d_in[7] (slot_mapping) unused: the new token is substituted in-kernel at
  // logical position context_len-1 (block tables are disjoint per sequence).
  (void)hipFuncSetAttribute((const void*)paged_attn_decode,
                            hipFuncAttributeMaxDynamicSharedMemorySize,
                            LDS_BYTES);
  dim3 grid(NUM_KV_HEADS, NUM_SEQS);
  paged_attn_decode<<<grid, dim3(THREADS), (size_t)LDS_BYTES, stream>>>(
      q, k, v, kc, vc, bt, cl, (float*)d_out);
}